// FourierMMLayer_52527450030329
// MI455X (gfx1250) — compile-verified
//
#include <hip/hip_runtime.h>
#include <hip/hip_bf16.h>

// FNet-style 2D DFT as 4 real f32 GEMMs on CDNA5 WMMA (v_wmma_f32_16x16x4_f32).
//   out = C*(x*C) - S*(x*S),  C[j,k]=cos(2*pi*j*k/N), S[j,k]=sin(2*pi*j*k/N), N=1024

#define SEQ   1024
#define HID   1024
#define BATCH 4

typedef float v2f __attribute__((ext_vector_type(2)));
typedef float v4f __attribute__((ext_vector_type(4)));
typedef float v8f __attribute__((ext_vector_type(8)));

// ---------------------------------------------------------------------------
// Twiddle generation: C[j,k]=cos(2*pi*(j*k mod N)/N), S[j,k]=sin(...)
// ---------------------------------------------------------------------------
__global__ __launch_bounds__(256) void fnet_twiddle_kernel(float* __restrict__ Cmat,
                                                           float* __restrict__ Smat) {
    const unsigned idx = blockIdx.x * 256u + threadIdx.x;   // 0 .. N*N-1
    const unsigned j = idx >> 10;
    const unsigned k = idx & 1023u;
    const unsigned m = (j * k) & 1023u;                     // j*k mod 1024
    const float theta = (float)m * (6.2831853071795864769f / 1024.0f);
    float s, c;
    __sincosf(theta, &s, &c);
    Cmat[idx] = c;
    Smat[idx] = s;
}

// ---------------------------------------------------------------------------
// Generic row-major f32 GEMM:  D = alpha * A(MxK) * B(KxN)  [+ D if accumulate]
// 128 threads = 4 waves; wave tile 64x64 (4x4 WMMA 16x16 accum); block 128x128.
// K staged through LDS in slabs of 32. B slab stored K-pair-interleaved so a
// B fragment is a single aligned ds_load_b64 into the WMMA source pair.
// ---------------------------------------------------------------------------
#define BK 32

__global__ __launch_bounds__(128) void fnet_gemm_f32_wmma(
    const float* __restrict__ A, const float* __restrict__ B, float* __restrict__ D,
    int M, int N, int K,
    long long sAb, long long sBb, long long sDb,
    float alpha, int accumulate)
{
    // A slab: row-major [128][BK] padded (stride 36 floats = 144B, 16B aligned)
    __shared__ float As[128][BK + 4];
    // B slab: pair-interleaved: Bs2[p][col] = { B[2p][col], B[2p+1][col] }
    // row stride 130 v2f = 1040B (16B aligned)
    __shared__ v2f Bs2[BK / 2][128 + 2];

    const int bz = blockIdx.z;
    A += (long long)bz * sAb;
    B += (long long)bz * sBb;
    D += (long long)bz * sDb;

    const int tid   = threadIdx.x;
    const int lane  = tid & 31;
    const int wave  = tid >> 5;            // 0..3
    const int waveM = wave >> 1;           // 0..1  -> 64-row slab
    const int waveN = wave & 1;            // 0..1  -> 64-col slab
    const int half  = lane >> 4;           // ISA lane-half (K split / M split)
    const int r     = lane & 15;

    const int rowBase = blockIdx.y * 128;
    const int colBase = blockIdx.x * 128;

    v8f acc[4][4];
#pragma unroll
    for (int i = 0; i < 4; ++i)
#pragma unroll
        for (int j = 0; j < 4; ++j)
            acc[i][j] = (v8f)0.0f;

    // global->LDS copy mapping
    const int atx = tid & 7;               // float4 column within A slab row
    const int aty = tid >> 3;              // 16 rows per sweep
    const int btx = tid & 31;              // float4 column group for B
    const int bty = tid >> 5;              // pair-row per sweep (0..3)

    for (int k0 = 0; k0 < K; k0 += BK) {
#pragma unroll
        for (int i = 0; i < 8; ++i) {      // A: 128x32 tile
            const int row = aty + 16 * i;
            const v4f v = *(const v4f*)(A + (long long)(rowBase + row) * K + k0 + 4 * atx);
            *(v4f*)&As[row][4 * atx] = v;
        }
#pragma unroll
        for (int i = 0; i < 4; ++i) {      // B: 32x128 tile -> 16 pair-rows
            const int p = bty + 4 * i;     // pair-row index (K rows 2p, 2p+1)
            const float* g0 = B + (long long)(k0 + 2 * p + 0) * N + colBase + 4 * btx;
            const float* g1 = B + (long long)(k0 + 2 * p + 1) * N + colBase + 4 * btx;
            const v4f a = *(const v4f*)g0;
            const v4f b = *(const v4f*)g1;
            v4f lo, hi;                    // interleave {a.x,b.x,a.y,b.y} {a.z,b.z,a.w,b.w}
            lo.x = a.x; lo.y = b.x; lo.z = a.y; lo.w = b.y;
            hi.x = a.z; hi.y = b.z; hi.z = a.w; hi.w = b.w;
            *(v4f*)&Bs2[p][4 * btx + 0] = lo;
            *(v4f*)&Bs2[p][4 * btx + 2] = hi;
        }
        __syncthreads();

#pragma unroll
        for (int kk = 0; kk < BK; kk += 4) {
            // A fragment (16x4 f32): v0 = K{0,2}, v1 = K{1,3} across lane halves
            v2f afrag[4], bfrag[4];
#pragma unroll
            for (int i = 0; i < 4; ++i) {
                const int row = waveM * 64 + i * 16 + r;
                afrag[i] = *(const v2f*)&As[row][kk + 2 * half];
            }
#pragma unroll
            for (int j = 0; j < 4; ++j) {
                const int col = waveN * 64 + j * 16 + r;
                bfrag[j] = Bs2[kk / 2 + half][col];   // single aligned b64 load
            }
#pragma unroll
            for (int i = 0; i < 4; ++i)
#pragma unroll
                for (int j = 0; j < 4; ++j)
                    acc[i][j] = __builtin_amdgcn_wmma_f32_16x16x4_f32(
                        /*neg_a=*/false, afrag[i],
                        /*neg_b=*/false, bfrag[j],
                        /*c_mod=*/(short)0, acc[i][j],
                        /*reuse_a=*/false, /*reuse_b=*/false);
        }
        __syncthreads();
    }

    // C/D layout: VGPR e holds M = e (lanes 0-15) / M = 8+e (lanes 16-31), N = r
#pragma unroll
    for (int i = 0; i < 4; ++i) {
        const int rowT = rowBase + waveM * 64 + i * 16 + 8 * half;
#pragma unroll
        for (int j = 0; j < 4; ++j) {
            const int col = colBase + waveN * 64 + j * 16 + r;
#pragma unroll
            for (int e = 0; e < 8; ++e) {
                float* p = D + (long long)(rowT + e) * N + col;
                const float v = alpha * acc[i][j][e];
                if (accumulate) *p += v; else *p = v;
            }
        }
    }
}

// ---------------------------------------------------------------------------
// Launch:  twiddles -> T1 = x*C, T2 = x*S -> out = C*T1 - S*T2
// ---------------------------------------------------------------------------
extern "C" void kernel_launch(void* const* d_in, const int* in_sizes, int n_in,
                              void* d_out, int out_size, void* d_ws, size_t ws_size,
                              hipStream_t stream) {
    const float* x   = (const float*)d_in[0];        // [B, S, D] f32
    float*       out = (float*)d_out;                // [B, S, D] f32

    // workspace layout (40 MB): C | S | T1 | T2
    float* Cmat = (float*)d_ws;
    float* Smat = Cmat + (size_t)SEQ * HID;
    float* T1   = Smat + (size_t)SEQ * HID;
    float* T2   = T1   + (size_t)BATCH * SEQ * HID;

    fnet_twiddle_kernel<<<(SEQ * HID) / 256, 256, 0, stream>>>(Cmat, Smat);

    const long long bstride = (long long)SEQ * HID;

    // Stage 1: hidden-axis DFT, batch folded into M: [B*S,1024] @ [1024,1024]
    dim3 g1(HID / 128, (BATCH * SEQ) / 128, 1);
    fnet_gemm_f32_wmma<<<g1, 128, 0, stream>>>(x, Cmat, T1,
        BATCH * SEQ, HID, HID, 0, 0, 0, 1.0f, 0);
    fnet_gemm_f32_wmma<<<g1, 128, 0, stream>>>(x, Smat, T2,
        BATCH * SEQ, HID, HID, 0, 0, 0, 1.0f, 0);

    // Stage 2: seq-axis DFT per batch: out[b] = C@T1[b] - S@T2[b]
    dim3 g2(HID / 128, SEQ / 128, BATCH);
    fnet_gemm_f32_wmma<<<g2, 128, 0, stream>>>(Cmat, T1, out,
        SEQ, HID, SEQ, 0, bstride, bstride, 1.0f, 0);
    fnet_gemm_f32_wmma<<<g2, 128, 0, stream>>>(Smat, T2, out,
        SEQ, HID, SEQ, 0, bstride, bstride, -1.0f, 1);
}